// MultiColorTransformLayer_55963423867505
// MI455X (gfx1250) — compile-verified
//
#include <hip/hip_runtime.h>

typedef __attribute__((ext_vector_type(2))) float v2f;
typedef __attribute__((ext_vector_type(8))) float v8f;

#define TPB 256
#define WAVES_PER_BLOCK (TPB / 32)

// out[p,d] = sum_c x[p,c] * w[c,d]; 16 pixels per wave per WMMA.
// A (16x4 f32): lane m (0-15): VGPR0=K0, VGPR1=K1 ; lane m+16: VGPR0=K2, VGPR1=K3(=0 pad)
// B (4x16 f32): VGPR0: lanes0-15 = row K0, lanes16-31 = row K2 ; VGPR1: K1 / K3(=0)
// C (16x16 f32): VGPR r: lanes0-15 = row r, lanes16-31 = row r+8
__global__ __launch_bounds__(TPB) void color3x3_wmma(const float* __restrict__ x,
                                                     const float* __restrict__ w,
                                                     float* __restrict__ out,
                                                     int numGroups) {
  __shared__ float4 ldsIn4[WAVES_PER_BLOCK][12];
  __shared__ float4 ldsOut4[WAVES_PER_BLOCK][12];

  const int lane        = threadIdx.x & 31;
  const int waveInBlock = threadIdx.x >> 5;
  const int waveId      = (int)((blockIdx.x * TPB + threadIdx.x) >> 5);
  const int wavesTotal  = (int)((gridDim.x * TPB) >> 5);

  const int n    = lane & 15;   // A: row m / B,C: column N
  const int half = lane >> 4;

  // Build B = w padded to 4x16 (rows K=c, cols N=d). Uniform per lane, loaded once.
  v2f B;
  B.x = 0.0f;
  B.y = 0.0f;
  if (n < 3) {
    B.x = w[(half ? 2 : 0) * 3 + n];        // row K = 0 (low lanes) / K = 2 (high lanes)
    B.y = half ? 0.0f : w[3 + n];           // row K = 1 (low lanes) / K = 3 -> zero pad
  }

  float*       mIn  = (float*)ldsIn4[waveInBlock];
  float*       mOut = (float*)ldsOut4[waveInBlock];
  const float4* x4  = (const float4*)x;

  for (int g = waveId; g < numGroups; g += wavesTotal) {
    const size_t base4 = (size_t)g * 12;    // 48 floats = 12 float4 per 16-pixel group

    // Stage 192 contiguous bytes into this wave's LDS slab: one b128 per lane (12 lanes).
    if (lane < 12) {
      ldsIn4[waveInBlock][lane] = x4[base4 + lane];
      // Hide latency of the next group's tile.
      if (g + wavesTotal < numGroups)
        __builtin_prefetch(&x4[base4 + (size_t)wavesTotal * 12 + lane], 0, 3);
    }
    // Same-wave LDS is processed in order; this waits out the staging writes and
    // fences the compiler so the cross-lane reads below aren't hoisted.
    asm volatile("s_wait_dscnt 0" ::: "memory");

    // Gather A from LDS per the 16x4 layout (pixel m = row, channel = K).
    v2f A;
    A.x = mIn[3 * n + (half ? 2 : 0)];
    A.y = half ? 0.0f : mIn[3 * n + 1];

    v8f C = {0.f, 0.f, 0.f, 0.f, 0.f, 0.f, 0.f, 0.f};
    C = __builtin_amdgcn_wmma_f32_16x16x4_f32(
        /*neg_a=*/false, A, /*neg_b=*/false, B,
        /*c_mod=*/(short)0, C, /*reuse_a=*/false, /*reuse_b=*/false);

    // Compact useful C columns (N=0..2) into LDS: lane n<3 holds out[row, n].
    if (n < 3) {
      const int rb = half * 8;              // rows r (low lanes) / r+8 (high lanes)
      mOut[(rb + 0) * 3 + n] = C[0];
      mOut[(rb + 1) * 3 + n] = C[1];
      mOut[(rb + 2) * 3 + n] = C[2];
      mOut[(rb + 3) * 3 + n] = C[3];
      mOut[(rb + 4) * 3 + n] = C[4];
      mOut[(rb + 5) * 3 + n] = C[5];
      mOut[(rb + 6) * 3 + n] = C[6];
      mOut[(rb + 7) * 3 + n] = C[7];
    }
    asm volatile("s_wait_dscnt 0" ::: "memory");

    // One coalesced b128 store per lane (12 lanes) -> 192 contiguous bytes.
    if (lane < 12) {
      ((float4*)out)[base4 + lane] = ldsOut4[waveInBlock][lane];
    }
  }
}

// Scalar tail for pixel counts not divisible by 16 (empty for the reference shape).
__global__ void color3x3_tail(const float* __restrict__ x, const float* __restrict__ w,
                              float* __restrict__ out, long start, long totalPixels) {
  long p = start + (long)blockIdx.x * blockDim.x + threadIdx.x;
  if (p >= totalPixels) return;
  const float c0 = x[3 * p + 0], c1 = x[3 * p + 1], c2 = x[3 * p + 2];
#pragma unroll
  for (int d = 0; d < 3; ++d)
    out[3 * p + d] = c0 * w[d] + c1 * w[3 + d] + c2 * w[6 + d];
}

extern "C" void kernel_launch(void* const* d_in, const int* in_sizes, int n_in,
                              void* d_out, int out_size, void* d_ws, size_t ws_size,
                              hipStream_t stream) {
  const float* x = (const float*)d_in[0];   // [32,512,512,3] f32
  const float* w = (const float*)d_in[1];   // [3,3] f32
  float* out = (float*)d_out;               // [32,512,512,3] f32

  const long totalPixels = (long)in_sizes[0] / 3;
  const long groups      = totalPixels / 16;     // 16 pixels per WMMA group

  if (groups > 0) {
    int blocks = 4096;
    long waves = (long)blocks * (TPB / 32);
    if (waves > groups) blocks = (int)((groups * 32 + TPB - 1) / TPB) + 1;
    color3x3_wmma<<<blocks, TPB, 0, stream>>>(x, w, out, (int)groups);
  }

  const long tailStart = groups * 16;
  const long tail      = totalPixels - tailStart;
  if (tail > 0) {
    int tb = (int)((tail + TPB - 1) / TPB);
    color3x3_tail<<<tb, TPB, 0, stream>>>(x, w, out, tailStart, totalPixels);
  }
}